// Embedding_Table_Cache_Group_10342281249258
// MI455X (gfx1250) — compile-verified
//
#include <hip/hip_runtime.h>

// Problem constants (from reference): T tables, B bags (1 idx/bag), D embed dim
#define TBL    4
#define BATCH  4096
#define DIM    64
#define CSET   100003
#define WAYS   4
#define AUX    4096
#define NROWS  500000
#define CACHE_ROWS (WAYS * CSET + AUX)   // rows in cache_weights per table

// ---------------------------------------------------------------------------
// Kernel A: per-table cache probe + sequential miss ranking (block scan).
// One block per table, 1024 threads (32 waves -> good latency hiding), each
// thread owns 4 CONSECUTIVE indices so the cumsum order matches the
// reference's sequential miss ranking exactly.
//   - indices loaded as one int4 (global_load_b128)
//   - 4-way occupancy probe is one int4 load (occ[set][0..3] is 16 B)
//   - results staged in LDS as int4 (ds_store_b128)
// Outputs:
//   out_cache_idx[t*B+n] = cache_lookup (hit row, or C*W + miss_rank)
//   ws_src[t*B+n]        = hit ? cache_row : -(idx+1)   (gather source code)
// ---------------------------------------------------------------------------
__global__ __launch_bounds__(1024)
void classify_scan_kernel(const int* __restrict__ lS_i,
                          const int* __restrict__ occ,
                          int* __restrict__ out_cache_idx,
                          int* __restrict__ ws_src)
{
    const int t   = blockIdx.x;
    const int tid = threadIdx.x;

    __shared__ int s_look[BATCH];   // hit_lookup (>=0) or -(idx+1) (<0)
    __shared__ int s_sum[1024];     // per-thread miss counts -> inclusive scan

    const int4* idx4 = (const int4*)(lS_i + t * BATCH);
    const int4* occ4 = (const int4*)(occ + (size_t)t * CSET * WAYS);

    // ---- probe: 4 consecutive indices per thread -------------------------
    const int4 iv = idx4[tid];
    int idxk[4] = { iv.x, iv.y, iv.z, iv.w };

    int setk[4];
    #pragma unroll
    for (int k = 0; k < 4; ++k) setk[k] = idxk[k] % CSET;

    int4 waysk[4];
    #pragma unroll
    for (int k = 0; k < 4; ++k) waysk[k] = occ4[setk[k]];   // 4 independent b128 loads

    int miss_cnt = 0;
    int4 lookv;
    int* lookp = (int*)&lookv;
    #pragma unroll
    for (int k = 0; k < 4; ++k) {
        const int idx = idxk[k];
        const int4 w = waysk[k];
        int way = -1;
        if      (w.x == idx) way = 0;    // first matching way (argmax semantics)
        else if (w.y == idx) way = 1;
        else if (w.z == idx) way = 2;
        else if (w.w == idx) way = 3;
        if (way >= 0) {
            lookp[k] = CSET * way + setk[k];   // resident cache row
        } else {
            lookp[k] = -idx - 1;               // miss marker: full-table row
            ++miss_cnt;
        }
    }
    ((int4*)s_look)[tid] = lookv;              // ds_store_b128

    // ---- inclusive Hillis-Steele scan of per-thread miss counts ----------
    s_sum[tid] = miss_cnt;
    __syncthreads();
    #pragma unroll 1
    for (int off = 1; off < 1024; off <<= 1) {
        int v = (tid >= off) ? s_sum[tid - off] : 0;
        __syncthreads();
        s_sum[tid] += v;
        __syncthreads();
    }
    int rank = s_sum[tid] - miss_cnt;          // misses strictly before my chunk

    // ---- emit lookups + gather sources (vector stores) -------------------
    const int4 lv = ((const int4*)s_look)[tid];
    const int lk[4] = { lv.x, lv.y, lv.z, lv.w };
    int4 outv;
    int* outp = (int*)&outv;
    #pragma unroll
    for (int k = 0; k < 4; ++k) {
        const int v = lk[k];
        if (v >= 0) {
            outp[k] = v;
        } else {
            outp[k] = CSET * WAYS + rank;      // packed into aux region in order
            ++rank;
        }
    }
    ((int4*)(out_cache_idx + t * BATCH))[tid] = outv;   // global_store_b128
    ((int4*)(ws_src        + t * BATCH))[tid] = lv;     // global_store_b128
}

// ---------------------------------------------------------------------------
// Kernel B: gather + EmbeddingBag(sum) pooling.
// 16 lanes per bag; each lane async-copies 16 B of the 256 B row from global
// memory straight into LDS (global_load_async_to_lds_b128, GV mode, tracked
// by ASYNCcnt), waits, then accumulates a float4 from LDS. Rows within a bag
// (generally one, since offsets == arange) are summed.
// ---------------------------------------------------------------------------
__global__ __launch_bounds__(256)
void gather_pool_kernel(const int* __restrict__ lS_o,
                        const int* __restrict__ ws_src,
                        const float* __restrict__ cw,
                        const float* __restrict__ ft,
                        float* __restrict__ out)
{
    __shared__ float stage[256 * 4];        // one float4 slot per thread

    const int tid  = threadIdx.x;
    const int grp  = tid >> 4;              // bag group within block (0..15)
    const int lane = tid & 15;              // 16 lanes x float4 = 64 floats
    const int g    = blockIdx.x * 16 + grp; // global bag id in [0, T*B)
    const int t    = g >> 12;               // g / BATCH (BATCH = 4096)
    const int b    = g & (BATCH - 1);

    const int* offs = lS_o + t * BATCH;
    const int start = offs[b];
    const int end   = (b + 1 < BATCH) ? offs[b + 1] : BATCH;

    // Low 32 bits of the generic LDS address == LDS byte offset (flat LDS
    // aperture places the offset in addr[31:0]).
    const unsigned ldsoff = (unsigned)(size_t)(&stage[tid * 4]);

    float4 acc = make_float4(0.f, 0.f, 0.f, 0.f);
    #pragma unroll 1
    for (int j = start; j < end; ++j) {
        const int s = ws_src[t * BATCH + j];
        const float* row = (s >= 0)
            ? cw + ((size_t)t * CACHE_ROWS + (size_t)s) * DIM
            : ft + ((size_t)t * NROWS + (size_t)(-s - 1)) * DIM;
        const unsigned long long ga = (unsigned long long)(row + lane * 4);

        // Async DMA 16 B global -> LDS (ASYNCcnt), then wait for arrival.
        asm volatile("global_load_async_to_lds_b128 %0, %1, off"
                     :: "v"(ldsoff), "v"(ga) : "memory");
        asm volatile("s_wait_asynccnt 0" ::: "memory");

        const float4 v = *reinterpret_cast<const float4*>(&stage[tid * 4]);
        acc.x += v.x; acc.y += v.y; acc.z += v.z; acc.w += v.w;
    }

    float4* o = reinterpret_cast<float4*>(out);
    o[(size_t)g * (DIM / 4) + lane] = acc;
}

// ---------------------------------------------------------------------------
// Host launcher. Inputs (setup_inputs order):
//   d_in[0]=lS_o [T,B] i32, d_in[1]=lS_i [T,B] i32, d_in[2]=occupancy [T,C,W] i32,
//   d_in[3]=cache_weights [T,W*C+A,D] f32, d_in[4]=full_tables [T,NROWS,D] f32
// Output: ly [T,B,D] f32 followed by cache_idxs [T,B] i32 (flat concatenation).
// ---------------------------------------------------------------------------
extern "C" void kernel_launch(void* const* d_in, const int* in_sizes, int n_in,
                              void* d_out, int out_size, void* d_ws, size_t ws_size,
                              hipStream_t stream)
{
    (void)in_sizes; (void)n_in; (void)out_size; (void)ws_size;

    const int*   lS_o = (const int*)d_in[0];
    const int*   lS_i = (const int*)d_in[1];
    const int*   occ  = (const int*)d_in[2];
    const float* cw   = (const float*)d_in[3];
    const float* ft   = (const float*)d_in[4];

    float* out_ly = (float*)d_out;
    int*   out_ci = (int*)(out_ly + (size_t)TBL * BATCH * DIM);
    int*   ws_src = (int*)d_ws;    // T*B ints = 64 KB scratch

    classify_scan_kernel<<<TBL, 1024, 0, stream>>>(lS_i, occ, out_ci, ws_src);
    gather_pool_kernel<<<(TBL * BATCH) / 16, 256, 0, stream>>>(lS_o, ws_src, cw, ft, out_ly);
}